// MultiHeadAttention_33947421508073
// MI455X (gfx1250) — compile-verified
//
#include <hip/hip_runtime.h>
#include <math.h>
#include <stdint.h>

typedef __attribute__((ext_vector_type(2))) float v2f;
typedef __attribute__((ext_vector_type(8))) float v8f;

#define HSZ   1024
#define NHEAD 16
#define HD    64
#define BATCH 2
#define SEQ   2048
#define MROWS (BATCH*SEQ)   // 4096

#define BM 64
#define BN 64
#define BK 32
#define KLDS (BK+4)   // 36: rows 16B-aligned for async b128, even for b64 frags, bank-spread
#define TLDS (HD+4)   // 68
#define NTHREADS 128  // 4 waves of 32

// ---------------------------------------------------------------------------
// CDNA5 async global->LDS copy (ASYNCcnt path). LDS dst = wave-relative byte
// offset (low 32 bits of the generic shared pointer); global src in VGPR pair.
// ---------------------------------------------------------------------------
__device__ __forceinline__ void async_b128(const float* lds_dst, const float* gsrc) {
    uint32_t dst = (uint32_t)(uintptr_t)lds_dst;
    asm volatile("global_load_async_to_lds_b128 %0, %1, off"
                 :: "v"(dst), "v"(gsrc) : "memory");
}
__device__ __forceinline__ void wait_async() {
    asm volatile("s_wait_asynccnt 0x0" ::: "memory");
}

// ---------------------------------------------------------------------------
// Fused QKV projection: out = x @ W^T + b, written in (B, NHEAD, SEQ, HD).
// grid = (HSZ/BN, MROWS/BM, 3); z selects q/k/v. Double-buffered async tiles.
// ---------------------------------------------------------------------------
__global__ __launch_bounds__(NTHREADS)
void mha_qkv_proj(const float* __restrict__ x,
                  const float* __restrict__ w0, const float* __restrict__ w1,
                  const float* __restrict__ w2,
                  const float* __restrict__ b0, const float* __restrict__ b1,
                  const float* __restrict__ b2,
                  float* __restrict__ o0, float* __restrict__ o1,
                  float* __restrict__ o2)
{
    __shared__ float Xs[2][BM*KLDS];
    __shared__ float Ws[2][BN*KLDS];

    const int which = blockIdx.z;
    const float* w    = which == 0 ? w0 : (which == 1 ? w1 : w2);
    const float* bias = which == 0 ? b0 : (which == 1 ? b1 : b2);
    float*       out  = which == 0 ? o0 : (which == 1 ? o1 : o2);

    const int tid  = threadIdx.x;
    const int wave = tid >> 5, lane = tid & 31, half = lane >> 4, ln = lane & 15;
    const int m0 = blockIdx.y * BM;
    const int n0 = blockIdx.x * BN;

    auto issue_tiles = [&](int kb, int buf) {
        for (int c = tid; c < BM*BK/4; c += NTHREADS) {      // X tile (BM x BK)
            int r = c >> 3, c4 = (c & 7) << 2;
            async_b128(&Xs[buf][r*KLDS + c4], x + (size_t)(m0 + r)*HSZ + kb + c4);
        }
        for (int c = tid; c < BN*BK/4; c += NTHREADS) {      // W tile (BN x BK), W is (N,K)
            int r = c >> 3, c4 = (c & 7) << 2;
            async_b128(&Ws[buf][r*KLDS + c4], w + (size_t)(n0 + r)*HSZ + kb + c4);
        }
    };

    v8f acc[4] = {};

    issue_tiles(0, 0);
    for (int kb = 0; kb < HSZ; kb += BK) {
        const int cur = (kb >> 5) & 1;
        wait_async();                       // this wave's copies for buf[cur] landed
        __syncthreads();                    // tile visible; prior reads of buf[cur^1] done
        if (kb + BK < HSZ) issue_tiles(kb + BK, cur ^ 1);   // overlap with compute

        #pragma unroll
        for (int ks = 0; ks < BK; ks += 4) {
            const int kh = ks + 2*half;
            v2f a = *(const v2f*)&Xs[cur][(wave*16 + ln)*KLDS + kh];   // A: 16x4, row=ln
            #pragma unroll
            for (int nt = 0; nt < 4; ++nt) {
                v2f b = *(const v2f*)&Ws[cur][(nt*16 + ln)*KLDS + kh]; // B[k][n]=W[n][k]
                acc[nt] = __builtin_amdgcn_wmma_f32_16x16x4_f32(
                    false, a, false, b, (short)0, acc[nt], false, false);
            }
        }
    }

    // epilogue -> (B, NHEAD, SEQ, HD)
    #pragma unroll
    for (int nt = 0; nt < 4; ++nt) {
        const int n = n0 + nt*16 + ln;
        const float bv = bias[n];
        const int h = n >> 6, d = n & 63;
        #pragma unroll
        for (int r = 0; r < 8; ++r) {
            const int m = m0 + wave*16 + r + 8*half;
            const int b = m >> 11, s = m & 2047;
            out[((size_t)((b*NHEAD + h)*SEQ + s))*HD + d] = acc[nt][r] + bv;
        }
    }
}

// ---------------------------------------------------------------------------
// Flash attention (causal), fp32 WMMA, one 64-row query tile per block.
// grid = (SEQ/BM, BATCH*NHEAD). Q/K/V/O in (B, NHEAD, SEQ, HD).
// K/V tiles double-buffered via async copies.
// ---------------------------------------------------------------------------
__global__ __launch_bounds__(NTHREADS)
void mha_flash_attn(const float* __restrict__ q, const float* __restrict__ k,
                    const float* __restrict__ v, float* __restrict__ o)
{
    __shared__ float Qs[BM*TLDS];
    __shared__ float Ps[BM*TLDS];
    __shared__ float Ks[2][BM*TLDS];
    __shared__ float Vs[2][BM*TLDS];

    const int tid  = threadIdx.x;
    const int wave = tid >> 5, lane = tid & 31, half = lane >> 4, ln = lane & 15;
    const int m0 = blockIdx.x * BM;
    const size_t base = (size_t)blockIdx.y * SEQ * HD;

    auto issue_kv = [&](int t0, int buf) {
        for (int c = tid; c < BM*HD/4; c += NTHREADS) {
            int r = c >> 4, c4 = (c & 15) << 2;
            async_b128(&Ks[buf][r*TLDS + c4], k + base + (size_t)(t0 + r)*HD + c4);
            async_b128(&Vs[buf][r*TLDS + c4], v + base + (size_t)(t0 + r)*HD + c4);
        }
    };

    for (int c = tid; c < BM*HD/4; c += NTHREADS) {          // Q tile, loaded once
        int r = c >> 4, c4 = (c & 15) << 2;
        async_b128(&Qs[r*TLDS + c4], q + base + (size_t)(m0 + r)*HD + c4);
    }
    issue_kv(0, 0);
    wait_async();
    __syncthreads();

    float mrow[8], lrow[8];
    v8f oacc[4] = {};
    #pragma unroll
    for (int r = 0; r < 8; ++r) { mrow[r] = -INFINITY; lrow[r] = 0.f; }

    const int ntiles = blockIdx.x + 1;                       // causal: skip fully-masked tiles
    for (int it = 0; it < ntiles; ++it) {
        const int cur = it & 1;
        const int t0 = it * BM;
        if (it + 1 < ntiles) issue_kv((it + 1)*BM, cur ^ 1); // overlap with compute

        // S = Q K^T for this wave's 16x64 strip (contraction over d)
        v8f sacc[4] = {};
        #pragma unroll
        for (int ks = 0; ks < HD; ks += 4) {
            const int kh = ks + 2*half;
            v2f a = *(const v2f*)&Qs[(wave*16 + ln)*TLDS + kh];
            #pragma unroll
            for (int nt = 0; nt < 4; ++nt) {
                v2f b = *(const v2f*)&Ks[cur][(nt*16 + ln)*TLDS + kh]; // B[d][t]=K[t][d]
                sacc[nt] = __builtin_amdgcn_wmma_f32_16x16x4_f32(
                    false, a, false, b, (short)0, sacc[nt], false, false);
            }
        }

        // Online softmax. C-tile row M=r+8*half lives across the lane's 16-lane
        // half -> per-lane row stats, shfl_xor(1,2,4,8) stays inside the half.
        #pragma unroll
        for (int r = 0; r < 8; ++r) {
            const int srow = m0 + wave*16 + r + 8*half;
            float mx = -INFINITY;
            #pragma unroll
            for (int nt = 0; nt < 4; ++nt) {
                const int scol = t0 + nt*16 + ln;
                float sv = sacc[nt][r] * 0.125f;             // 1/sqrt(64)
                if (scol > srow) sv = -INFINITY;             // causal mask
                sacc[nt][r] = sv;
                mx = fmaxf(mx, sv);
            }
            #pragma unroll
            for (int off = 8; off > 0; off >>= 1)
                mx = fmaxf(mx, __shfl_xor(mx, off, 32));
            const float mnew = fmaxf(mrow[r], mx);           // finite: diag always valid
            const float corr = __expf(mrow[r] - mnew);
            mrow[r] = mnew;
            float rsum = 0.f;
            #pragma unroll
            for (int nt = 0; nt < 4; ++nt) {
                const float p = __expf(sacc[nt][r] - mnew);
                sacc[nt][r] = p;
                rsum += p;
            }
            #pragma unroll
            for (int off = 8; off > 0; off >>= 1)
                rsum += __shfl_xor(rsum, off, 32);
            lrow[r] = lrow[r] * corr + rsum;
            #pragma unroll
            for (int dt = 0; dt < 4; ++dt)
                oacc[dt][r] = oacc[dt][r] * corr;
        }

        // Spill P (C-layout) to LDS so it can be re-read in A-layout.
        // Each wave touches only its own 16 rows -> no barrier needed.
        #pragma unroll
        for (int r = 0; r < 8; ++r) {
            const int prow = wave*16 + r + 8*half;
            #pragma unroll
            for (int nt = 0; nt < 4; ++nt)
                Ps[prow*TLDS + nt*16 + ln] = sacc[nt][r];
        }

        // O += P @ V (contraction over the 64 keys of this tile)
        #pragma unroll
        for (int ts = 0; ts < BM; ts += 4) {
            const int th = ts + 2*half;
            v2f a = *(const v2f*)&Ps[(wave*16 + ln)*TLDS + th];
            #pragma unroll
            for (int dt = 0; dt < 4; ++dt) {
                v2f b;                                        // B[t][d] = Vs row-major
                b.x = Vs[cur][(th    )*TLDS + dt*16 + ln];
                b.y = Vs[cur][(th + 1)*TLDS + dt*16 + ln];
                oacc[dt] = __builtin_amdgcn_wmma_f32_16x16x4_f32(
                    false, a, false, b, (short)0, oacc[dt], false, false);
            }
        }

        wait_async();        // next tile's async copies landed (no-op on last iter)
        __syncthreads();     // all waves done reading buf[cur]; safe to overwrite next iter
    }

    #pragma unroll
    for (int dt = 0; dt < 4; ++dt) {
        const int d = dt*16 + ln;
        #pragma unroll
        for (int r = 0; r < 8; ++r) {
            const int srow = m0 + wave*16 + r + 8*half;
            o[base + (size_t)srow*HD + d] = oacc[dt][r] / lrow[r];
        }
    }
}

// ---------------------------------------------------------------------------
// Output projection: out = attn @ Wo^T + bo; attn gathered from (B,H,S,D)
// back to (B,S,H*D) during the async LDS fill. grid = (HSZ/BN, MROWS/BM).
// ---------------------------------------------------------------------------
__global__ __launch_bounds__(NTHREADS)
void mha_out_proj(const float* __restrict__ attn, const float* __restrict__ w,
                  const float* __restrict__ bias, float* __restrict__ out)
{
    __shared__ float Xs[2][BM*KLDS];
    __shared__ float Ws[2][BN*KLDS];

    const int tid  = threadIdx.x;
    const int wave = tid >> 5, lane = tid & 31, half = lane >> 4, ln = lane & 15;
    const int m0 = blockIdx.y * BM;
    const int n0 = blockIdx.x * BN;

    auto issue_tiles = [&](int kb, int buf) {
        for (int c = tid; c < BM*BK/4; c += NTHREADS) {
            int r = c >> 3, c4 = (c & 7) << 2;
            const int m = m0 + r, b = m >> 11, s = m & 2047;
            const int kcol = kb + c4, h = kcol >> 6, d = kcol & 63;
            async_b128(&Xs[buf][r*KLDS + c4],
                       attn + ((size_t)((b*NHEAD + h)*SEQ + s))*HD + d);
        }
        for (int c = tid; c < BN*BK/4; c += NTHREADS) {
            int r = c >> 3, c4 = (c & 7) << 2;
            async_b128(&Ws[buf][r*KLDS + c4], w + (size_t)(n0 + r)*HSZ + kb + c4);
        }
    };

    v8f acc[4] = {};

    issue_tiles(0, 0);
    for (int kb = 0; kb < HSZ; kb += BK) {
        const int cur = (kb >> 5) & 1;
        wait_async();
        __syncthreads();
        if (kb + BK < HSZ) issue_tiles(kb + BK, cur ^ 1);

        #pragma unroll
        for (int ks = 0; ks < BK; ks += 4) {
            const int kh = ks + 2*half;
            v2f a = *(const v2f*)&Xs[cur][(wave*16 + ln)*KLDS + kh];
            #pragma unroll
            for (int nt = 0; nt < 4; ++nt) {
                v2f b = *(const v2f*)&Ws[cur][(nt*16 + ln)*KLDS + kh];
                acc[nt] = __builtin_amdgcn_wmma_f32_16x16x4_f32(
                    false, a, false, b, (short)0, acc[nt], false, false);
            }
        }
    }

    #pragma unroll
    for (int nt = 0; nt < 4; ++nt) {
        const int n = n0 + nt*16 + ln;
        const float bv = bias[n];
        #pragma unroll
        for (int r = 0; r < 8; ++r) {
            const int m = m0 + wave*16 + r + 8*half;
            out[(size_t)m*HSZ + n] = acc[nt][r] + bv;
        }
    }
}

// ---------------------------------------------------------------------------
extern "C" void kernel_launch(void* const* d_in, const int* in_sizes, int n_in,
                              void* d_out, int out_size, void* d_ws, size_t ws_size,
                              hipStream_t stream) {
    (void)in_sizes; (void)n_in; (void)out_size; (void)ws_size;
    const float* x  = (const float*)d_in[0];
    // d_in[1] = attention_mask: causal tril mask, applied analytically in-kernel
    const float* wq = (const float*)d_in[2];
    const float* bq = (const float*)d_in[3];
    const float* wk = (const float*)d_in[4];
    const float* bk = (const float*)d_in[5];
    const float* wv = (const float*)d_in[6];
    const float* bv = (const float*)d_in[7];
    const float* wo = (const float*)d_in[8];
    const float* bo = (const float*)d_in[9];
    float* out = (float*)d_out;

    const size_t per = (size_t)BATCH * NHEAD * SEQ * HD;     // 4Mi floats = 16MB
    float* Q = (float*)d_ws;                                 // ws usage: 64MB total
    float* K = Q + per;
    float* V = K + per;
    float* A = V + per;

    dim3 blk(NTHREADS);
    hipLaunchKernelGGL(mha_qkv_proj, dim3(HSZ/BN, MROWS/BM, 3), blk, 0, stream,
                       x, wq, wk, wv, bq, bk, bv, Q, K, V);
    hipLaunchKernelGGL(mha_flash_attn, dim3(SEQ/BM, BATCH*NHEAD), blk, 0, stream,
                       Q, K, V, A);
    hipLaunchKernelGGL(mha_out_proj, dim3(HSZ/BN, MROWS/BM), blk, 0, stream,
                       A, wo, bo, out);
}